// MetaFeatureExtractor_42502996361644
// MI455X (gfx1250) — compile-verified
//
#include <hip/hip_runtime.h>
#include <math.h>

#define NBANDS 34
#define HW     4096
#define BATCH  256

typedef float v2f __attribute__((ext_vector_type(2)));
typedef float v8f __attribute__((ext_vector_type(8)));

__device__ __forceinline__ int clamp64(int i) { return i < 0 ? 0 : (i > 63 ? 63 : i); }
__device__ __forceinline__ float zpix(const float* img, int r, int c) {
  return (r >= 0 && r < 64 && c >= 0 && c < 64) ? img[r * 64 + c] : 0.0f;
}

// ---------------------------------------------------------------------------
// K0: generate the 16 Gabor filters into workspace (max 19x19 taps each).
// Order: th_idx*4 + sg_idx*2 + f_idx  (matches reference list comprehension)
// ---------------------------------------------------------------------------
__global__ void gen_gabor_kernel(float* __restrict__ filt) {
  const int fi = blockIdx.x;
  const float PI = 3.14159265358979f;
  float th = (float)(fi >> 2) * (PI * 0.25f);
  float sg = ((fi >> 1) & 1) ? 3.0f : 1.0f;
  float fr = (fi & 1) ? 0.3f : 0.1f;
  float ct = cosf(th), st = sinf(th);
  float m = fmaxf(fmaxf(fabsf(3.0f * sg * ct), fabsf(3.0f * sg * st)), 1.0f);
  int rad = (int)ceilf(m);
  int K = 2 * rad + 1;
  float inv2s2 = 0.5f / (sg * sg);
  float norm = 1.0f / (2.0f * PI * sg * sg);
  for (int idx = threadIdx.x; idx < K * K; idx += blockDim.x) {
    int yy = idx / K - rad;
    int xx = idx % K - rad;
    float rotx =  (float)xx * ct + (float)yy * st;
    float roty = -(float)xx * st + (float)yy * ct;
    float g = expf(-(rotx * rotx + roty * roty) * inv2s2) * norm;
    filt[fi * 368 + idx] = g * cosf(2.0f * PI * fr * rotx);
  }
}

// ---------------------------------------------------------------------------
// K1: one streaming pass over x[b]: channel means, global mean/var,
// channel-diff variance, spatial mean image.
// ---------------------------------------------------------------------------
__global__ void stats_kernel(const float* __restrict__ x,
                             float* __restrict__ spatial,
                             float* __restrict__ ms,
                             float* __restrict__ scal) {
  const int b = blockIdx.x;
  const int tid = threadIdx.x;  // 256 threads
  __shared__ float sChan[NBANDS];
  __shared__ float sSS, sDS, sDSS;
  if (tid < NBANDS) sChan[tid] = 0.f;
  if (tid == 0) { sSS = 0.f; sDS = 0.f; sDSS = 0.f; }
  __syncthreads();

  const float* xb = x + (size_t)b * NBANDS * HW;
  float sp[16], prev[16];
  float ss = 0.f, dS = 0.f, dSS = 0.f;
  for (int c = 0; c < NBANDS; ++c) {
    const float* xc = xb + c * HW;
    float cs = 0.f;
#pragma unroll
    for (int i = 0; i < 16; ++i) {
      float v = xc[tid + i * 256];
      cs += v;
      ss += v * v;
      if (c == 0) {
        sp[i] = v;
      } else {
        sp[i] += v;
        float dd = v - prev[i];
        dS += dd; dSS += dd * dd;
      }
      prev[i] = v;
    }
    atomicAdd(&sChan[c], cs);
  }
  const float invC = 1.0f / (float)NBANDS;
#pragma unroll
  for (int i = 0; i < 16; ++i)
    spatial[(size_t)b * HW + tid + i * 256] = sp[i] * invC;
  atomicAdd(&sSS, ss); atomicAdd(&sDS, dS); atomicAdd(&sDSS, dSS);
  __syncthreads();

  if (tid < NBANDS) ms[b * 40 + tid] = sChan[tid] * (1.0f / (float)HW);
  if (tid == 0) {
    float tot = 0.f;
    for (int c = 0; c < NBANDS; ++c) tot += sChan[c];
    const float N = (float)(NBANDS * HW);
    float mu = tot / N;
    float var = sSS / N - mu * mu;
    float sd = sqrtf(fmaxf(var, 0.f));
    const float Nd = (float)((NBANDS - 1) * HW);
    float md = sDS / Nd;
    float sgv = sDSS / Nd - md * md;
    float* s = scal + b * 16;
    s[0] = mu; s[1] = sd; s[2] = sgv;
  }
}

// ---------------------------------------------------------------------------
// K2: features on the 64x64 spatial image (in LDS): box counts, contrast,
// Sobel edge fraction, LBP variance, 16 Gabor conv variances.
// ---------------------------------------------------------------------------
__global__ void spatial_kernel(const float* __restrict__ spatial,
                               const float* __restrict__ filt,
                               float* __restrict__ scal) {
  const int b = blockIdx.x;
  const int tid = threadIdx.x;  // 256 threads
  __shared__ float sImg[HW];
  __shared__ float sQ[HW];
  __shared__ float bmx[1364], bmn[1364];
  __shared__ float gS[16], gSS[16];
  __shared__ float sContr, sLbpS, sLbpSS;
  __shared__ int iCnt[6];

  if (tid < 16) { gS[tid] = 0.f; gSS[tid] = 0.f; }
  if (tid == 0) { sContr = 0.f; sLbpS = 0.f; sLbpSS = 0.f; }
  if (tid < 6) iCnt[tid] = 0;

  const float* spb = spatial + (size_t)b * HW;
#pragma unroll
  for (int i = 0; i < 16; ++i) {
    int p = tid + i * 256;
    float v = spb[p];
    sImg[p] = v;
    sQ[p] = floorf(v * 255.0f);
  }
  __syncthreads();

  // ---- contrast: mean of squared horizontal diffs of q ----
  float csum = 0.f;
#pragma unroll
  for (int i = 0; i < 16; ++i) {
    int p = tid + i * 256, c = p & 63;
    if (c < 63) { float d = sQ[p] - sQ[p + 1]; csum += d * d; }
  }
  atomicAdd(&sContr, csum);

  // ---- box counting pyramid (s = 2,4,8,16,32) ----
  for (int cell = tid; cell < 1024; cell += 256) {
    int i2 = cell >> 5, j2 = cell & 31;
    int base = (i2 * 2) * 64 + j2 * 2;
    float a = sQ[base], bq = sQ[base + 1], cq = sQ[base + 64], dq = sQ[base + 65];
    float mx = fmaxf(fmaxf(a, bq), fmaxf(cq, dq));
    float mn = fminf(fminf(a, bq), fminf(cq, dq));
    bmx[cell] = mx; bmn[cell] = mn;
    if (mx > mn) atomicAdd(&iCnt[0], 1);
  }
  __syncthreads();
  {
    int inOff = 0, inW = 32;
    const int outOffs[4] = {1024, 1280, 1344, 1360};
    for (int lev = 0; lev < 4; ++lev) {
      int outW = inW >> 1;
      int outOff = outOffs[lev];
      for (int cell = tid; cell < outW * outW; cell += 256) {
        int i2 = cell / outW, j2 = cell % outW;
        int i0 = inOff + (i2 * 2) * inW + j2 * 2;
        float mx = fmaxf(fmaxf(bmx[i0], bmx[i0 + 1]), fmaxf(bmx[i0 + inW], bmx[i0 + inW + 1]));
        float mn = fminf(fminf(bmn[i0], bmn[i0 + 1]), fminf(bmn[i0 + inW], bmn[i0 + inW + 1]));
        bmx[outOff + cell] = mx; bmn[outOff + cell] = mn;
        if (mx > mn) atomicAdd(&iCnt[lev + 1], 1);
      }
      __syncthreads();
      inOff = outOff; inW = outW;
    }
  }

  // ---- Sobel edge (symmetric pad => clamp) ----
  int ecnt = 0;
#pragma unroll
  for (int i = 0; i < 16; ++i) {
    int p = tid + i * 256, r = p >> 6, c = p & 63;
    int rm = clamp64(r - 1), rp = clamp64(r + 1);
    int cm = clamp64(c - 1), cp = clamp64(c + 1);
    float v00 = sImg[rm * 64 + cm], v01 = sImg[rm * 64 + c], v02 = sImg[rm * 64 + cp];
    float v10 = sImg[r  * 64 + cm],                            v12 = sImg[r  * 64 + cp];
    float v20 = sImg[rp * 64 + cm], v21 = sImg[rp * 64 + c], v22 = sImg[rp * 64 + cp];
    float gx = (v02 + 2.f * v12 + v22) - (v00 + 2.f * v10 + v20);
    float gy = (v20 + 2.f * v21 + v22) - (v00 + 2.f * v01 + v02);
    if (gx * gx + gy * gy > 0.01f) ecnt++;
  }
  atomicAdd(&iCnt[5], ecnt);

  // ---- LBP variance (zero pad, bilinear taps) ----
  {
    const float SQ = 0.70710678f;
    const float offs[8][2] = {{0.f, 1.f}, {-SQ, SQ}, {-1.f, 0.f}, {-SQ, -SQ},
                              {0.f, -1.f}, {SQ, -SQ}, {1.f, 0.f}, {SQ, SQ}};
    float lsum = 0.f, lss = 0.f;
    for (int i = 0; i < 16; ++i) {
      int p = tid + i * 256, r = p >> 6, c = p & 63;
      float center = sImg[p];
      int bits[8];
#pragma unroll
      for (int k = 0; k < 8; ++k) {
        float dr = offs[k][0], dc = offs[k][1];
        int r0 = (int)floorf(dr), c0 = (int)floorf(dc);
        float tr = dr - (float)r0, tc = dc - (float)c0;
        float w00 = (1.f - tr) * (1.f - tc), w01 = (1.f - tr) * tc;
        float w10 = tr * (1.f - tc), w11 = tr * tc;
        float val = 0.f;
        if (w00 > 1e-12f) val += w00 * zpix(sImg, r + r0,     c + c0);
        if (w01 > 1e-12f) val += w01 * zpix(sImg, r + r0,     c + c0 + 1);
        if (w10 > 1e-12f) val += w10 * zpix(sImg, r + r0 + 1, c + c0);
        if (w11 > 1e-12f) val += w11 * zpix(sImg, r + r0 + 1, c + c0 + 1);
        bits[k] = (val >= center) ? 1 : 0;
      }
      int trans = 0, ones = 0;
#pragma unroll
      for (int k = 0; k < 8; ++k) { trans += (bits[k] != bits[(k + 1) & 7]); ones += bits[k]; }
      float lbp = (trans <= 2) ? (float)ones : 9.0f;
      lsum += lbp; lss += lbp * lbp;
    }
    atomicAdd(&sLbpS, lsum); atomicAdd(&sLbpSS, lss);
  }

  // ---- Gabor conv variances (coeffs are wave-uniform loads) ----
  {
    const float PI = 3.14159265358979f;
    for (int fi = 0; fi < 16; ++fi) {
      float th = (float)(fi >> 2) * (PI * 0.25f);
      float sg = ((fi >> 1) & 1) ? 3.0f : 1.0f;
      float ct = cosf(th), st = sinf(th);
      float m = fmaxf(fmaxf(fabsf(3.0f * sg * ct), fabsf(3.0f * sg * st)), 1.0f);
      int rad = (int)ceilf(m);
      int K = 2 * rad + 1;
      float acc[16];
#pragma unroll
      for (int i = 0; i < 16; ++i) acc[i] = 0.f;
      const float* fbase = filt + fi * 368;
      for (int u = -rad; u <= rad; ++u)
        for (int v = -rad; v <= rad; ++v) {
          float w = fbase[(u + rad) * K + (v + rad)];
#pragma unroll
          for (int i = 0; i < 16; ++i) {
            int p = tid + i * 256, r = p >> 6, c = p & 63;
            int rr = r + u, cc = c + v;
            float val = (rr >= 0 && rr < 64 && cc >= 0 && cc < 64) ? sImg[rr * 64 + cc] : 0.f;
            acc[i] += w * val;
          }
        }
      float s1 = 0.f, s2 = 0.f;
#pragma unroll
      for (int i = 0; i < 16; ++i) { s1 += acc[i]; s2 += acc[i] * acc[i]; }
      atomicAdd(&gS[fi], s1); atomicAdd(&gSS[fi], s2);
    }
  }
  __syncthreads();

  if (tid == 0) {
    float* s = scal + b * 16;
    s[3] = sContr / (64.0f * 63.0f);
    for (int l = 0; l < 5; ++l) s[4 + l] = fmaxf((float)iCnt[l], 1.0f);
    s[9] = (float)iCnt[5] / 4096.0f;
    float lm = sLbpS / 4096.0f;
    float lvar = sLbpSS / 4096.0f - lm * lm;
    s[10] = fminf(1.0f, lvar / 100.0f);
    float et = 0.f;
    for (int fi = 0; fi < 16; ++fi) {
      float gm = gS[fi] / 4096.0f;
      et += gSS[fi] / 4096.0f - gm * gm;
    }
    s[11] = et / 16.0f / 100.0f;
  }
}

// ---------------------------------------------------------------------------
// K3: per-batch Gram matrix G = X X^T (34x34, padded to 48x48) via
// V_WMMA_F32_16X16X4_F32.  One wave per (upper-triangle tile pair, batch):
// 6 pairs instead of 9, off-diagonal tiles stored mirrored (G symmetric).
// Channel padding needs NO masking: padded rows (>=34) produce garbage in
// Gram rows/cols >= 34 which the final kernel never reads; addresses are
// clamped so loads stay in-bounds.  Inner loop is pure b64-load + wmma.
// A layout (16x4 f32): lane L, vgpr v holds  M = L%16, K = 2*(L/16)+v.
// B layout (4x16 f32): lane L, vgpr v holds  N = L%16, K = 2*(L/16)+v (B=X^T).
// ---------------------------------------------------------------------------
__global__ void gram_wmma_kernel(const float* __restrict__ x,
                                 float* __restrict__ gram) {
  static const int PCI[6] = {0, 0, 0, 1, 1, 2};
  static const int PCJ[6] = {0, 1, 2, 1, 2, 2};
  const int pair = blockIdx.x;   // 0..5
  const int b = blockIdx.y;
  const int ci = PCI[pair], cj = PCJ[pair];
  const int lane = threadIdx.x;  // 32 threads, EXEC all ones for WMMA
  const int mrow = lane & 15;
  const int koff = (lane >> 4) * 2;
  const int rowA = ci * 16 + mrow;
  const int rowB = cj * 16 + mrow;
  const int rA = rowA < NBANDS ? rowA : NBANDS - 1;  // clamp: stay in-bounds
  const int rB = rowB < NBANDS ? rowB : NBANDS - 1;
  const float* xb = x + (size_t)b * NBANDS * HW;
  const float* pa = xb + (size_t)rA * HW + koff;
  const float* pb = xb + (size_t)rB * HW + koff;

  v8f acc = {0.f, 0.f, 0.f, 0.f, 0.f, 0.f, 0.f, 0.f};
  for (int k = 0; k < HW; k += 4) {
    v2f a = *(const v2f*)(pa + k);
    v2f bb = *(const v2f*)(pb + k);
    acc = __builtin_amdgcn_wmma_f32_16x16x4_f32(
        /*neg_a=*/false, a, /*neg_b=*/false, bb,
        /*c_mod=*/(short)0, acc, /*reuse_a=*/false, /*reuse_b=*/false);
  }

  // D layout 16x16 f32: lane L: N = L%16; vgpr r: M = r + 8*(L/16)
  float* gb = gram + (size_t)b * 48 * 48;
  const int n = lane & 15;
  const int mbase = ci * 16 + 8 * (lane >> 4);
  const int ncol = cj * 16 + n;
#pragma unroll
  for (int r = 0; r < 8; ++r)
    gb[(mbase + r) * 48 + ncol] = acc[r];
  if (ci != cj) {  // mirror into the lower-triangle tile
#pragma unroll
    for (int r = 0; r < 8; ++r)
      gb[ncol * 48 + (mbase + r)] = acc[r];
  }
}

// ---------------------------------------------------------------------------
// K4: per-batch feature assembly (one thread per batch element).
// ---------------------------------------------------------------------------
__global__ void final_kernel(const float* __restrict__ ms,
                             const float* __restrict__ gram,
                             const float* __restrict__ scal,
                             float* __restrict__ out) {
  const int b = blockIdx.x * blockDim.x + threadIdx.x;
  if (b >= BATCH) return;
  const float* m = ms + b * 40;
  const float* g = gram + (size_t)b * 48 * 48;
  const float* s = scal + b * 16;

  float mv[NBANDS];
  for (int i = 0; i < NBANDS; ++i) mv[i] = m[i];

  // average pairwise correlation: cov_ij = G_ij - P * mu_i * mu_j
  float d[NBANDS];
  for (int i = 0; i < NBANDS; ++i) {
    float cii = g[i * 48 + i] - (float)HW * mv[i] * mv[i];
    d[i] = sqrtf(fmaxf(cii, 0.f));
  }
  float csum = 0.f;
  for (int i = 0; i < NBANDS; ++i)
    for (int j = 0; j < NBANDS; ++j) {
      if (i == j) continue;
      float cij = g[i * 48 + j] - (float)HW * mv[i] * mv[j];
      csum += cij / (d[i] * d[j]);
    }
  float avg_corr = csum / (float)(NBANDS * (NBANDS - 1));

  // spectral entropy + contrast
  float mn = mv[0], mx = mv[0];
  for (int i = 1; i < NBANDS; ++i) { mn = fminf(mn, mv[i]); mx = fmaxf(mx, mv[i]); }
  float tot = 0.f;
  for (int i = 0; i < NBANDS; ++i) tot += (mv[i] - mn);
  float hl = 0.f;
  for (int i = 0; i < NBANDS; ++i) {
    float p = (mv[i] - mn) / (tot + 1e-8f);
    hl -= p * log2f(p + 1e-8f);
  }
  hl /= log2f((float)NBANDS);
  float sc = mx - mn;

  // absorption depth (peak prominence on f = -ms)
  float f[NBANDS];
  for (int i = 0; i < NBANDS; ++i) f[i] = -mv[i];
  float totp = 0.f;
  int cnt = 0;
  for (int i = 1; i < NBANDS - 1; ++i) {
    if (!(f[i] > f[i - 1] && f[i] > f[i + 1])) continue;
    int lb = -1, rb = NBANDS;
    for (int j = 0; j < i; ++j) if (f[j] > f[i]) lb = j;              // nearest left higher (max j)
    for (int j = NBANDS - 1; j > i; --j) if (f[j] > f[i]) rb = j;     // nearest right higher (min j)
    float lmin = __builtin_inff(), rmin = __builtin_inff();
    for (int j = lb + 1; j < i; ++j) lmin = fminf(lmin, f[j]);
    for (int j = i + 1; j < rb; ++j) rmin = fminf(rmin, f[j]);
    float prom = f[i] - fmaxf(lmin, rmin);
    if (prom >= 0.05f) { cnt++; totp += prom; }
  }
  float ad = (cnt > 0) ? totp / (float)cnt : 0.f;

  // fractal slope from box counts
  const float LN2 = 0.6931471805599453f;
  float slope = 0.f;
  for (int l = 0; l < 5; ++l) {
    float lxc = ((float)l - 2.0f) * LN2;  // lx = (l+1)ln2, mean = 3 ln2
    slope += logf(s[4 + l]) * lxc;
  }
  slope /= (10.0f * LN2 * LN2);
  float df = fminf(fmaxf(-slope, 1.0f), 2.0f);

  float mu = s[0], sd = s[1];
  float snr = 20.0f * log10f(mu / (sd + 1e-8f));
  snr = fminf(fmaxf(snr, 0.f), 50.f) / 50.f;
  float hs = 1.0f / (1.0f + s[3] / 100.0f);

  float feats[12] = {df, s[2], snr, hs, hl, mu, avg_corr, s[11], s[9], sc, s[10], ad};
  float* o = out + b * 12;
  for (int k = 0; k < 12; ++k) {
    float fv = fminf(fmaxf(feats[k], 0.f), 1.f);
    o[k] = 1.0f / (1.0f + expf(-fv));
  }
}

// ---------------------------------------------------------------------------
extern "C" void kernel_launch(void* const* d_in, const int* in_sizes, int n_in,
                              void* d_out, int out_size, void* d_ws, size_t ws_size,
                              hipStream_t stream) {
  (void)in_sizes; (void)n_in; (void)out_size; (void)ws_size;
  const float* x = (const float*)d_in[0];
  float* out = (float*)d_out;
  float* ws = (float*)d_ws;
  // workspace layout (floats)
  float* spatial = ws;                       // 256*4096       = 1,048,576
  float* ms      = spatial + BATCH * HW;     // 256*40         =    10,240
  float* gram    = ms + BATCH * 40;          // 256*48*48      =   589,824
  float* scal    = gram + BATCH * 48 * 48;   // 256*16         =     4,096
  float* filt    = scal + BATCH * 16;        // 16*368         =     5,888
  // total ~6.6 MB

  gen_gabor_kernel<<<16, 64, 0, stream>>>(filt);
  stats_kernel<<<BATCH, 256, 0, stream>>>(x, spatial, ms, scal);
  spatial_kernel<<<BATCH, 256, 0, stream>>>(spatial, filt, scal);
  gram_wmma_kernel<<<dim3(6, BATCH), 32, 0, stream>>>(x, gram);
  final_kernel<<<1, BATCH, 0, stream>>>(ms, gram, scal, out);
}